// Loss_yolov3_35201551958296
// MI455X (gfx1250) — compile-verified
//
#include <hip/hip_runtime.h>
#include <hip/hip_bf16.h>

#define NB 150          // ground-truth boxes per batch
#define NC 80           // classes
#define YOLO_B 16       // batch size
#define EPS 1e-6f

typedef __attribute__((ext_vector_type(2))) float v2f;
typedef __attribute__((ext_vector_type(8))) float v8f;

__device__ __forceinline__ float bce_logits(float x, float t) {
    // max(x,0) - x*t + log1p(exp(-|x|))
    return fmaxf(x, 0.0f) - x * t + log1pf(__expf(-fabsf(x)));
}

__global__ __launch_bounds__(256)
void yolo_layer_kernel(const float* __restrict__ p,
                       const float* __restrict__ pd,
                       const float* __restrict__ label,
                       const float* __restrict__ bboxes,
                       int posPerBatch,           // G*G*A
                       int wsBlockOffset,         // block-partial offset in ws (in triples)
                       float* __restrict__ ws)
{
    __shared__ float bb_x1[NB], bb_y1[NB], bb_x2[NB], bb_y2[NB], bb_a[NB];
    __shared__ float waveSums[8][3];

    const int b   = blockIdx.y;
    const int tpb = gridDim.x * 256;               // threads covering one batch

    // Stage this batch's GT boxes to LDS as xyxy + area
    for (int i = threadIdx.x; i < NB; i += 256) {
        const float* src = bboxes + ((long long)b * NB + i) * 4;
        float cx = src[0], cy = src[1], w = src[2], h = src[3];
        float ax = cx - 0.5f * w, bx = cx + 0.5f * w;
        float ay = cy - 0.5f * h, by = cy + 0.5f * h;
        float x1 = fminf(ax, bx), x2 = fmaxf(ax, bx);
        float y1 = fminf(ay, by), y2 = fmaxf(ay, by);
        bb_x1[i] = x1; bb_y1[i] = y1; bb_x2[i] = x2; bb_y2[i] = y2;
        bb_a[i]  = (x2 - x1) * (y2 - y1);
    }
    __syncthreads();

    float gAcc = 0.0f, cAcc = 0.0f, kAcc = 0.0f;

    const int iters = (posPerBatch + tpb - 1) / tpb;   // uniform across block
    for (int it = 0; it < iters; ++it) {
        int pos = it * tpb + blockIdx.x * 256 + threadIdx.x;
        if (pos < posPerBatch) {
            long long gpos = (long long)b * posPerBatch + pos;

            // pred box -> xyxy
            const float* pdp = pd + gpos * 4;
            float pcx = pdp[0], pcy = pdp[1], pw = pdp[2], ph = pdp[3];
            float ax = pcx - 0.5f * pw, bx = pcx + 0.5f * pw;
            float ay = pcy - 0.5f * ph, by = pcy + 0.5f * ph;
            float plx = fminf(ax, bx), phx = fmaxf(ax, bx);
            float ply = fminf(ay, by), phy = fmaxf(ay, by);
            float parea = (phx - plx) * (phy - ply);

            const float* lbp = label + gpos * 86;
            float obj = lbp[4];
            float mix = lbp[5];
            const float* pp = p + gpos * 85;
            float conf = pp[4];

            // Obj positions (~2%) will need 160 scattered floats after the scan:
            // prefetch those cachelines now so they arrive under the IoU loop
            // (gfx1250 global_prefetch_b8).
            bool hasObj = obj > 0.0f;
            if (hasObj) {
                __builtin_prefetch(pp + 5,  0, 3);
                __builtin_prefetch(pp + 37, 0, 3);
                __builtin_prefetch(pp + 69, 0, 3);
                __builtin_prefetch(lbp + 6,  0, 3);
                __builtin_prefetch(lbp + 38, 0, 3);
                __builtin_prefetch(lbp + 70, 0, 3);
            }

            // ---- max IoU vs 150 GT boxes (division-free running max) ----
            float bestI = 0.0f, bestU = 1.0f;   // represents IoU == 0
            #pragma unroll 5
            for (int i = 0; i < NB; ++i) {
                float lux = fmaxf(plx, bb_x1[i]);
                float luy = fmaxf(ply, bb_y1[i]);
                float rdx = fminf(phx, bb_x2[i]);
                float rdy = fminf(phy, bb_y2[i]);
                float wx  = fmaxf(rdx - lux, 0.0f);
                float wy  = fmaxf(rdy - luy, 0.0f);
                float inter = wx * wy;
                float uni   = parea + bb_a[i] - inter;
                // inter/(uni+eps) > bestI/(bestU+eps), denominators > 0
                bool gt = inter * (bestU + EPS) > bestI * (uni + EPS);
                bestI = gt ? inter : bestI;
                bestU = gt ? uni   : bestU;
            }
            float iouMax = bestI / (bestU + EPS);
            float noobj  = (1.0f - obj) * ((iouMax < 0.5f) ? 1.0f : 0.0f);

            // ---- focal objectness ----
            float sig   = 1.0f / (1.0f + __expf(-conf));
            float dcf   = fabsf(obj - sig);
            float focal = bce_logits(conf, obj) * (dcf * dcf);   // ALPHA=1, GAMMA=2
            cAcc += (obj + noobj) * focal * mix;

            // ---- obj-gated: GIoU + class BCE ----
            if (hasObj) {
                float lcx = lbp[0], lcy = lbp[1], lw = lbp[2], lh = lbp[3];
                float lax = lcx - 0.5f * lw, lbx2 = lcx + 0.5f * lw;
                float lay = lcy - 0.5f * lh, lby2 = lcy + 0.5f * lh;
                float llx = fminf(lax, lbx2), lhx = fmaxf(lax, lbx2);
                float lly = fminf(lay, lby2), lhy = fmaxf(lay, lby2);
                float larea = (lhx - llx) * (lhy - lly);

                float lux = fmaxf(plx, llx), luy = fmaxf(ply, lly);
                float rdx = fminf(phx, lhx), rdy = fminf(phy, lhy);
                float wx  = fmaxf(rdx - lux, 0.0f);
                float wy  = fmaxf(rdy - luy, 0.0f);
                float inter = wx * wy;
                float uni   = parea + larea - inter;
                float iou   = inter / (uni + EPS);

                float ex1 = fminf(plx, llx), ey1 = fminf(ply, lly);
                float ex2 = fmaxf(phx, lhx), ey2 = fmaxf(phy, lhy);
                float ew  = fmaxf(ex2 - ex1, 0.0f);
                float eh  = fmaxf(ey2 - ey1, 0.0f);
                float enc = ew * eh;
                float giou = iou - (enc - uni) / (enc + EPS);

                float scale = 2.0f - lw * lh * (1.0f / (416.0f * 416.0f));
                gAcc += obj * scale * (1.0f - giou) * mix;

                float ksum = 0.0f;
                #pragma unroll 4
                for (int c2 = 0; c2 < NC; ++c2)
                    ksum += bce_logits(pp[5 + c2], lbp[6 + c2]);
                kAcc += obj * ksum * mix;
            }
        }
    }

    // ---- wave reduction via V_WMMA_F32_16X16X4_F32 (EXEC all-ones here) ----
    // A(16x4): lane m -> A[m][0], lane m+16 -> A[m][2]; K=1,3 zero.
    // B(4x16) = ones  =>  D[m][n] = val(m) + val(m+16) for every n.
    v2f ones; ones[0] = 1.0f; ones[1] = 1.0f;
    float vals[3] = {gAcc, cAcc, kAcc};
    float totals[3];
    #pragma unroll
    for (int q = 0; q < 3; ++q) {
        v2f a; a[0] = vals[q]; a[1] = 0.0f;
        v8f c = {};
        c = __builtin_amdgcn_wmma_f32_16x16x4_f32(
                /*neg_a=*/false, a, /*neg_b=*/false, ones,
                /*c_mod=*/(short)0, c, /*reuse_a=*/false, /*reuse_b=*/false);
        float s = c[0] + c[1] + c[2] + c[3] + c[4] + c[5] + c[6] + c[7];
        // lanes 0-15 hold sum of rows 0..7, lanes 16-31 rows 8..15: add halves
        s += __shfl_down(s, 16, 32);
        totals[q] = s;   // lane 0 of each wave holds full 32-lane sum
    }

    int wave = threadIdx.x >> 5;
    int lane = threadIdx.x & 31;
    if (lane == 0) {
        waveSums[wave][0] = totals[0];
        waveSums[wave][1] = totals[1];
        waveSums[wave][2] = totals[2];
    }
    __syncthreads();
    if (threadIdx.x == 0) {
        float g = 0.0f, cc = 0.0f, k = 0.0f;
        #pragma unroll
        for (int w2 = 0; w2 < 8; ++w2) {
            g  += waveSums[w2][0];
            cc += waveSums[w2][1];
            k  += waveSums[w2][2];
        }
        float* dst = ws + (size_t)(wsBlockOffset + blockIdx.y * gridDim.x + blockIdx.x) * 3;
        dst[0] = g; dst[1] = cc; dst[2] = k;
    }
}

__global__ void yolo_finalize_kernel(const float* __restrict__ ws, int nBlocks,
                                     float* __restrict__ out)
{
    if (blockIdx.x == 0 && threadIdx.x == 0) {
        float g = 0.0f, c = 0.0f, k = 0.0f;
        for (int i = 0; i < nBlocks; ++i) {
            g += ws[i * 3 + 0];
            c += ws[i * 3 + 1];
            k += ws[i * 3 + 2];
        }
        const float invB = 1.0f / (float)YOLO_B;
        g *= invB; c *= invB; k *= invB;
        out[0] = g + c + k;   // total
        out[1] = g;           // giou
        out[2] = c;           // conf
        out[3] = k;           // cls
    }
}

extern "C" void kernel_launch(void* const* d_in, const int* in_sizes, int n_in,
                              void* d_out, int out_size, void* d_ws, size_t ws_size,
                              hipStream_t stream) {
    (void)n_in; (void)out_size; (void)ws_size;
    float* ws = (float*)d_ws;

    // dict order: p_s, pd_s, label_s, bboxes_s, p_m, pd_m, label_m, bboxes_m, p_l, pd_l, label_l, bboxes_l
    const int base[3] = {0, 4, 8};
    int off = 0;
    int posArr[3], bpbArr[3], offArr[3];
    for (int i = 0; i < 3; ++i) {
        int pos = in_sizes[base[i] + 1] / (YOLO_B * 4);   // pd flat count / (B*4)
        int bpb = (pos + 255) / 256;                      // ~1 position/thread -> 672 blocks total
        posArr[i] = pos; bpbArr[i] = bpb; offArr[i] = off;
        off += bpb * YOLO_B;
    }

    for (int i = 0; i < 3; ++i) {
        const float* p  = (const float*)d_in[base[i] + 0];
        const float* pd = (const float*)d_in[base[i] + 1];
        const float* lb = (const float*)d_in[base[i] + 2];
        const float* bb = (const float*)d_in[base[i] + 3];
        dim3 grid(bpbArr[i], YOLO_B);
        hipLaunchKernelGGL(yolo_layer_kernel, grid, dim3(256), 0, stream,
                           p, pd, lb, bb, posArr[i], offArr[i], ws);
    }
    hipLaunchKernelGGL(yolo_finalize_kernel, dim3(1), dim3(32), 0, stream,
                       ws, off, (float*)d_out);
}